// Conv2DSubPixel_44032004718623
// MI455X (gfx1250) — compile-verified
//
#include <hip/hip_runtime.h>

// Pixel-shuffle (depth-to-space), r=2:
//   in : [16, 256, 256, 64]  fp32
//   out: [16, 512, 512, 16]  fp32
//   out[b, x, y, c] = in[b, x>>1, y>>1, 32*(y&1) + 16*(x&1) + c]
//
// Pure permutation, 0 FLOPs, 537 MB of traffic -> HBM-bound (~23 us at
// 23.3 TB/s). The permutation is contiguous at 64B granularity, so the
// optimal path is direct global->global B128 copies with non-temporal
// cache hints. No WMMA / LDS / TDM: there is no arithmetic and no
// sub-64B reordering, so matrix pipes and staging would only add latency.

typedef float f4 __attribute__((ext_vector_type(4)));

__global__ __launch_bounds__(256) void pixel_shuffle_r2_kernel(
    const f4* __restrict__ in, f4* __restrict__ out, unsigned total)
{
    // One thread copies the (i=0, i=1) float4 pair for a given
    // (b, h, y, c4). The two input float4s are 64B apart, so each wave
    // consumes FULL 128B input cache lines; the two stores land in output
    // rows x=2h and x=2h+1, each fully coalesced (512B contiguous / wave
    // per store instruction).
    unsigned u = blockIdx.x * 256u + threadIdx.x;
    if (u >= total) return;

    // u = [b:4][h:8][y:9][c4:2]   (16 * 256 * 512 * 4 = 8,388,608 threads)
    unsigned c4 = u & 3u;          // which float4 of the 16-ch output pixel
    unsigned y  = (u >> 2) & 511u; // output column 0..511
    unsigned h  = (u >> 11) & 255u;// input row  (output rows 2h, 2h+1)
    unsigned b  = u >> 19;         // batch

    unsigned w = y >> 1;           // input column
    unsigned j = y & 1u;          // y parity

    // Input float4 index: pixel (b,h,w) has 16 float4s; chunk = 8*j + 4*i + c4
    unsigned pixIn  = (((b << 8) + h) << 8) + w;
    unsigned inIdx  = (pixIn << 4) + (j << 3) + c4;      // i = 0
    // Output float4 index: ((b*512 + x)*512 + y)*4 + c4, x = 2h
    unsigned outIdx = (((((b << 9) + (h << 1)) << 9) + y) << 2) + c4;

    // Non-temporal: every element moves exactly once -> keep L2 clean.
    f4 v0 = __builtin_nontemporal_load(&in[inIdx]);        // i = 0 half
    f4 v1 = __builtin_nontemporal_load(&in[inIdx + 4u]);   // i = 1 half (64B away)
    __builtin_nontemporal_store(v0, &out[outIdx]);          // row x = 2h
    __builtin_nontemporal_store(v1, &out[outIdx + 2048u]);  // row x = 2h+1 (+512*4 f4)
}

extern "C" void kernel_launch(void* const* d_in, const int* in_sizes, int n_in,
                              void* d_out, int out_size, void* d_ws, size_t ws_size,
                              hipStream_t stream) {
    (void)in_sizes; (void)n_in; (void)d_ws; (void)ws_size;
    const f4* in  = (const f4*)d_in[0];
    f4*       out = (f4*)d_out;

    // out_size = 16*512*512*16 = 67,108,864 floats.
    // Each thread moves 8 floats (two float4s) -> total threads = out_size/8.
    unsigned total  = (unsigned)(out_size / 8);           // 8,388,608
    unsigned blocks = (total + 255u) / 256u;              // 32768

    pixel_shuffle_r2_kernel<<<blocks, 256, 0, stream>>>(in, out, total);
}